// VariationalGCNEncoder_446676599435
// MI455X (gfx1250) — compile-verified
//
#include <hip/hip_runtime.h>
#include <hip/hip_bf16.h>

typedef __attribute__((ext_vector_type(2))) float v2f;
typedef __attribute__((ext_vector_type(8))) float v8f;

#define IN_CH   128
#define HID     128
#define OUT_CH  64

// ---------------------------------------------------------------------------
// Degree / normalization
// ---------------------------------------------------------------------------
__global__ void init_deg_kernel(float* __restrict__ deg, int n) {
    int i = blockIdx.x * blockDim.x + threadIdx.x;
    if (i < n) deg[i] = 1.0f;  // self-loop contributes 1 to every node
}

__global__ void accum_deg_kernel(const int* __restrict__ dst,
                                 float* __restrict__ deg, int e) {
    int i = blockIdx.x * blockDim.x + threadIdx.x;
    if (i < e) atomicAdd(&deg[dst[i]], 1.0f);
}

__global__ void dinv_kernel(float* __restrict__ deg, int n) {
    int i = blockIdx.x * blockDim.x + threadIdx.x;
    if (i < n) deg[i] = rsqrtf(deg[i]);  // deg >= 1 always (self-loops)
}

// ---------------------------------------------------------------------------
// f32 WMMA GEMM: C[M x 128] = A[M x 128] * W[128 x 128]
// 1 block = 128 threads = 4 waves; each block owns a 16-row M tile;
// wave w covers N columns [32w, 32w+32) as two 16x16 WMMA tiles.
// A layout (16x4 f32): lanes 0-15 K={0,1}, lanes 16-31 K={2,3}.
// B layout (4x16 f32): lanes = N column, VGPR pair = K rows (mirrored).
// C layout (16x16 f32): VGPR r: lanes 0-15 -> M=r, lanes 16-31 -> M=r+8.
// ---------------------------------------------------------------------------
__global__ __launch_bounds__(128)
void gemm_wmma_f32(const float* __restrict__ A, const float* __restrict__ W,
                   float* __restrict__ C, int M) {
    const int m0   = blockIdx.x * 16;
    if (m0 >= M) return;
    const int wave = threadIdx.x >> 5;
    const int lane = threadIdx.x & 31;
    const int l15  = lane & 15;
    const int kh   = (lane >> 4) << 1;     // 0 or 2
    const int nb   = wave * 32;            // base of this wave's 32 columns

    v8f acc0 = {};
    v8f acc1 = {};

    const int row = (m0 + l15 < M) ? (m0 + l15) : (M - 1);
    const float* __restrict__ arow = A + (size_t)row * IN_CH;
    const int n0 = nb + l15;
    const int n1 = nb + 16 + l15;
    const float* __restrict__ wc0 = W + n0;
    const float* __restrict__ wc1 = W + n1;

    for (int k = 0; k < IN_CH; k += 4) {
        const int kk = k + kh;
        v2f a = *(const v2f*)(arow + kk);

        v2f b0, b1;
        b0.x = wc0[(size_t)kk * HID];
        b0.y = wc0[(size_t)(kk + 1) * HID];
        b1.x = wc1[(size_t)kk * HID];
        b1.y = wc1[(size_t)(kk + 1) * HID];

        acc0 = __builtin_amdgcn_wmma_f32_16x16x4_f32(
                   false, a, false, b0, (short)0, acc0, false, false);
        acc1 = __builtin_amdgcn_wmma_f32_16x16x4_f32(
                   false, a, false, b1, (short)0, acc1, false, false);
    }

    const int mbase = m0 + ((lane >> 4) << 3);  // +0 or +8
    float* __restrict__ crow = C + (size_t)mbase * HID;
    if (m0 + 16 <= M) {
        // full tile: unconditional stores, no per-element exec masking
        #pragma unroll
        for (int r = 0; r < 8; ++r) {
            crow[(size_t)r * HID + n0] = acc0[r];
            crow[(size_t)r * HID + n1] = acc1[r];
        }
    } else {
        #pragma unroll
        for (int r = 0; r < 8; ++r) {
            if (mbase + r < M) {
                crow[(size_t)r * HID + n0] = acc0[r];
                crow[(size_t)r * HID + n1] = acc1[r];
            }
        }
    }
}

// ---------------------------------------------------------------------------
// Layer-1 aggregation: init (self-loop + bias), edge scatter, ReLU
// ---------------------------------------------------------------------------
__global__ void init_agg_l1(const float4* __restrict__ h0,
                            const float* __restrict__ dinv,
                            const float4* __restrict__ b1,
                            float4* __restrict__ h, int n) {
    int idx = blockIdx.x * blockDim.x + threadIdx.x;   // n * 32 float4's
    if (idx >= n * (HID / 4)) return;
    int node = idx >> 5, c4 = idx & 31;
    float w = dinv[node];
    w *= w;
    float4 v = h0[idx];
    float4 b = b1[c4];
    h[idx] = make_float4(v.x * w + b.x, v.y * w + b.y,
                         v.z * w + b.z, v.w * w + b.w);
}

// One edge per wave; each lane handles 4 consecutive channels (float4 gather,
// 4 coalesced f32 atomics). src/dst are wave-uniform -> scalarized.
__global__ __launch_bounds__(128)
void scatter_l1(const float* __restrict__ feat, const int* __restrict__ src,
                const int* __restrict__ dst, const float* __restrict__ dinv,
                float* __restrict__ out, int e) {
    const int wave = threadIdx.x >> 5;
    const int lane = threadIdx.x & 31;
    const int ed   = blockIdx.x * 4 + wave;
    if (ed >= e) return;
    const int s = src[ed], d = dst[ed];
    const float w = dinv[s] * dinv[d];
    const float4 v = *(const float4*)(feat + (size_t)s * HID + lane * 4);
    float* o = out + (size_t)d * HID + lane * 4;
    atomicAdd(o + 0, v.x * w);
    atomicAdd(o + 1, v.y * w);
    atomicAdd(o + 2, v.z * w);
    atomicAdd(o + 3, v.w * w);
}

__global__ void relu_kernel(float4* __restrict__ h, int total4) {
    int i = blockIdx.x * blockDim.x + threadIdx.x;
    if (i < total4) {
        float4 v = h[i];
        h[i] = make_float4(fmaxf(v.x, 0.0f), fmaxf(v.y, 0.0f),
                           fmaxf(v.z, 0.0f), fmaxf(v.w, 0.0f));
    }
}

// ---------------------------------------------------------------------------
// Concatenate W_mu|W_logstd into a 128x128 weight so both output heads run
// as a single WMMA GEMM.
// ---------------------------------------------------------------------------
__global__ void build_wcat(const float* __restrict__ wmu,
                           const float* __restrict__ wls,
                           float* __restrict__ wcat) {
    int idx = blockIdx.x * blockDim.x + threadIdx.x;   // 128*128
    if (idx >= HID * HID) return;
    int k = idx >> 7, n = idx & 127;
    wcat[idx] = (n < OUT_CH) ? wmu[k * OUT_CH + n]
                             : wls[k * OUT_CH + (n - OUT_CH)];
}

// ---------------------------------------------------------------------------
// Output aggregation: channels 0..63 -> mu, 64..127 -> logstd
// ---------------------------------------------------------------------------
__global__ void init_agg_out(const float4* __restrict__ t,
                             const float* __restrict__ dinv,
                             const float4* __restrict__ bmu,
                             const float4* __restrict__ bls,
                             float4* __restrict__ mu,
                             float4* __restrict__ logstd, int n) {
    int idx = blockIdx.x * blockDim.x + threadIdx.x;   // n * 32 float4's
    if (idx >= n * (HID / 4)) return;
    int node = idx >> 5, c4 = idx & 31;                // c4*4 = channel base
    float w = dinv[node];
    w *= w;
    float4 v = t[idx];
    if (c4 < OUT_CH / 4) {
        float4 b = bmu[c4];
        mu[(size_t)node * (OUT_CH / 4) + c4] =
            make_float4(v.x * w + b.x, v.y * w + b.y,
                        v.z * w + b.z, v.w * w + b.w);
    } else {
        int c = c4 - OUT_CH / 4;
        float4 b = bls[c];
        logstd[(size_t)node * (OUT_CH / 4) + c] =
            make_float4(v.x * w + b.x, v.y * w + b.y,
                        v.z * w + b.z, v.w * w + b.w);
    }
}

// One edge per wave; lane*4 channels never straddle the mu/logstd boundary
// (lanes 0-15 -> mu, 16-31 -> logstd), so the target select is one uniform
// pointer computation per lane.
__global__ __launch_bounds__(128)
void scatter_out(const float* __restrict__ t, const int* __restrict__ src,
                 const int* __restrict__ dst, const float* __restrict__ dinv,
                 float* __restrict__ mu, float* __restrict__ logstd, int e) {
    const int wave = threadIdx.x >> 5;
    const int lane = threadIdx.x & 31;
    const int ed   = blockIdx.x * 4 + wave;
    if (ed >= e) return;
    const int s = src[ed], d = dst[ed];
    const float w = dinv[s] * dinv[d];
    const int c0 = lane * 4;
    const float4 v = *(const float4*)(t + (size_t)s * HID + c0);
    float* o = (c0 < OUT_CH)
             ? mu     + (size_t)d * OUT_CH + c0
             : logstd + (size_t)d * OUT_CH + (c0 - OUT_CH);
    atomicAdd(o + 0, v.x * w);
    atomicAdd(o + 1, v.y * w);
    atomicAdd(o + 2, v.z * w);
    atomicAdd(o + 3, v.w * w);
}

// ---------------------------------------------------------------------------
// Launch
// ---------------------------------------------------------------------------
extern "C" void kernel_launch(void* const* d_in, const int* in_sizes, int n_in,
                              void* d_out, int out_size, void* d_ws, size_t ws_size,
                              hipStream_t stream) {
    const float* x       = (const float*)d_in[0];
    const int*   edges   = (const int*)d_in[1];
    const float* W1      = (const float*)d_in[2];
    const float* b1      = (const float*)d_in[3];
    const float* W_mu    = (const float*)d_in[4];
    const float* b_mu    = (const float*)d_in[5];
    const float* W_ls    = (const float*)d_in[6];
    const float* b_ls    = (const float*)d_in[7];

    const int N = in_sizes[0] / IN_CH;      // 100000
    const int E = in_sizes[1] / 2;          // 1600000
    const int* src = edges;
    const int* dst = edges + E;

    float* mu     = (float*)d_out;
    float* logstd = mu + (size_t)N * OUT_CH;

    // workspace layout (floats): dinv[N] | bufA[N*128] | bufB[N*128] | wcat[128*128]
    float* dinv = (float*)d_ws;
    float* bufA = dinv + (((size_t)N + 63) & ~(size_t)63);  // h0, then t
    float* bufB = bufA + (size_t)N * HID;                   // h
    float* wcat = bufB + (size_t)N * HID;

    const int total  = N * HID;
    const int total4 = total / 4;

    // 1) normalization
    init_deg_kernel<<<(N + 255) / 256, 256, 0, stream>>>(dinv, N);
    accum_deg_kernel<<<(E + 255) / 256, 256, 0, stream>>>(dst, dinv, E);
    dinv_kernel<<<(N + 255) / 256, 256, 0, stream>>>(dinv, N);

    // 2) h0 = x @ W1  (WMMA f32)
    gemm_wmma_f32<<<(N + 15) / 16, 128, 0, stream>>>(x, W1, bufA, N);

    // 3) layer-1 aggregation + ReLU
    init_agg_l1<<<(total4 + 255) / 256, 256, 0, stream>>>(
        (const float4*)bufA, dinv, (const float4*)b1, (float4*)bufB, N);
    scatter_l1<<<(E + 3) / 4, 128, 0, stream>>>(bufA, src, dst, dinv, bufB, E);
    relu_kernel<<<(total4 + 255) / 256, 256, 0, stream>>>((float4*)bufB, total4);

    // 4) t = h @ [W_mu | W_logstd]  (WMMA f32)
    build_wcat<<<(HID * HID + 255) / 256, 256, 0, stream>>>(W_mu, W_ls, wcat);
    gemm_wmma_f32<<<(N + 15) / 16, 128, 0, stream>>>(bufB, wcat, bufA, N);

    // 5) output aggregation (mu / logstd split)
    init_agg_out<<<(total4 + 255) / 256, 256, 0, stream>>>(
        (const float4*)bufA, dinv, (const float4*)b_mu, (const float4*)b_ls,
        (float4*)mu, (float4*)logstd, N);
    scatter_out<<<(E + 3) / 4, 128, 0, stream>>>(bufA, src, dst, dinv,
                                                 mu, logstd, E);
}